// AdjunctionModel_71803263255047
// MI455X (gfx1250) — compile-verified
//
#include <hip/hip_runtime.h>
#include <math.h>

typedef __attribute__((ext_vector_type(2))) float v2f;
typedef __attribute__((ext_vector_type(8))) float v8f;

// ---------------------------------------------------------------------------
// Pack Q into the exact V_WMMA_F32_16X16X4_F32 B-operand lane layout:
//   tile t, lane l (l=0..31), j=l&15, point q = Q[t*16+j]:
//     lanes 0..15  : (B[0][j], B[1][j]) = (-2qx, -2qy)
//     lanes 16..31 : (B[2][j], B[3][j]) = (-2qz, ||q||^2)
// Stored as float2 at pack[(b*nTiles + t)*32 + l] so the GEMM loop needs just
// one coalesced b64 load per lane per tile.
// ---------------------------------------------------------------------------
__global__ __launch_bounds__(128) void chamfer_pack(
    const float* __restrict__ Q,   // (B, NQ, 3)
    float* __restrict__ pack,      // (B, NQ/16, 32, 2)
    int NQ)
{
    const int gid = blockIdx.x * blockDim.x + threadIdx.x;  // over NQ*2 slots
    const int b   = blockIdx.y;
    if (gid >= NQ * 2) return;

    const int tile = gid >> 5;
    const int lane = gid & 31;
    const int j    = lane & 15;
    const int half = lane >> 4;
    const int n    = tile * 16 + j;

    const size_t qbase = ((size_t)b * NQ + n) * 3;
    const float qx = Q[qbase + 0], qy = Q[qbase + 1], qz = Q[qbase + 2];
    const float qsq = qx * qx + qy * qy + qz * qz;

    v2f bm;
    bm.x = half ? (-2.0f * qz) : (-2.0f * qx);
    bm.y = half ? qsq          : (-2.0f * qy);

    ((v2f*)pack)[((size_t)b * (NQ >> 4) + tile) * 32 + lane] = bm;
}

// ---------------------------------------------------------------------------
// One wave = one 16-row strip of P; loops over all pre-packed Q tiles.
// Per tile: 1 b64 load + 1 v_wmma_f32_16x16x4_f32 + running row mins.
// C[m][n] = -2 p.q + ||q||^2 ; dist = C + ||p||^2 added once after the min.
// ---------------------------------------------------------------------------
__global__ __launch_bounds__(128) void chamfer_dir_wmma(
    const float* __restrict__ P,     // (B, NP, 3) "row" points
    const float* __restrict__ packQ, // (B, NQ/16, 32, 2) packed B operands
    float* __restrict__ partial,     // (B, NP/16) strip partial means
    int NP, int NQ)
{
    const int lane        = threadIdx.x & 31;
    const int wave        = threadIdx.x >> 5;
    const int wavesPerBlk = blockDim.x >> 5;
    const int nStrips     = NP >> 4;
    const int nTiles      = NQ >> 4;
    const int strip       = blockIdx.x * wavesPerBlk + wave;
    const int b           = blockIdx.y;
    if (strip >= nStrips) return;   // wave-uniform: EXEC all-1s at every WMMA

    const int j    = lane & 15;
    const int half = lane >> 4;

    // A tile: row M=j of the strip, (px,py,pz,1) split across lane halves
    const size_t pbase = ((size_t)b * NP + (size_t)strip * 16 + j) * 3;
    const float px = P[pbase + 0], py = P[pbase + 1], pz = P[pbase + 2];
    const float psq = px * px + py * py + pz * pz;

    v2f a;
    a.x = half ? pz   : px;   // VGPR0: K=0 | K=2
    a.y = half ? 1.0f : py;   // VGPR1: K=1 | K=3

    float rmin[8];
#pragma unroll
    for (int v = 0; v < 8; ++v) rmin[v] = __builtin_inff();

    const v2f* qt = (const v2f*)packQ + (size_t)b * nTiles * 32 + lane;

#pragma unroll 4
    for (int t = 0; t < nTiles; ++t) {
        const v2f bm = qt[(size_t)t * 32];       // single coalesced b64 load
        v8f c = {};
        // 8 args: (neg_a, A, neg_b, B, c_mod, C, reuse_a, reuse_b)
        c = __builtin_amdgcn_wmma_f32_16x16x4_f32(
                false, a, false, bm, (short)0, c, false, false);
        // c[v] holds (M = v + 8*half, N = j)
#pragma unroll
        for (int v = 0; v < 8; ++v) rmin[v] = fminf(rmin[v], c[v]);
    }

    // Cross-lane min over the 16 column slots inside each half
#pragma unroll
    for (int v = 0; v < 8; ++v) {
        float r = rmin[v];
        r = fminf(r, __shfl_xor(r, 1, 32));
        r = fminf(r, __shfl_xor(r, 2, 32));
        r = fminf(r, __shfl_xor(r, 4, 32));
        r = fminf(r, __shfl_xor(r, 8, 32));
        rmin[v] = r;             // row min of C for row M = v + 8*half
    }

    // Sum 16 row mins across both halves
    float rsum = 0.0f;
#pragma unroll
    for (int v = 0; v < 8; ++v) rsum += rmin[v];
    rsum += __shfl_xor(rsum, 16, 32);

    // Sum of ||p||^2 over the strip (each half holds all 16 values)
    float ssum = psq;
    ssum += __shfl_xor(ssum, 1, 32);
    ssum += __shfl_xor(ssum, 2, 32);
    ssum += __shfl_xor(ssum, 4, 32);
    ssum += __shfl_xor(ssum, 8, 32);

    const float stripSum = rsum + ssum;          // sum_m min_n dist[m][n]
    if (lane == 0)
        partial[(size_t)b * nStrips + strip] = stripSum / (float)NP;
}

// Deterministic per-batch tree reduction of both directions' strip partials.
__global__ __launch_bounds__(256) void chamfer_reduce(
    const float* __restrict__ w1, int s1,
    const float* __restrict__ w2, int s2,
    float* __restrict__ out)
{
    __shared__ float sh[256];
    const int b = blockIdx.x, tid = threadIdx.x;
    float s = 0.0f;
    for (int i = tid; i < s1; i += blockDim.x) s += w1[(size_t)b * s1 + i];
    for (int i = tid; i < s2; i += blockDim.x) s += w2[(size_t)b * s2 + i];
    sh[tid] = s;
    __syncthreads();
    for (int off = 128; off > 0; off >>= 1) {
        if (tid < off) sh[tid] += sh[tid + off];
        __syncthreads();
    }
    if (tid == 0) out[b] = sh[0];
}

extern "C" void kernel_launch(void* const* d_in, const int* in_sizes, int n_in,
                              void* d_out, int out_size, void* d_ws, size_t ws_size,
                              hipStream_t stream) {
    const float* pc1 = (const float*)d_in[0];   // (B, N1, 3) f32
    const float* pc2 = (const float*)d_in[1];   // (B, N2, 3) f32
    float* out = (float*)d_out;                 // (B,) f32

    const int B  = out_size;
    const int N1 = in_sizes[0] / (3 * B);
    const int N2 = in_sizes[1] / (3 * B);
    const int s1 = N1 >> 4;                     // strips when rows = pc1
    const int s2 = N2 >> 4;                     // strips when rows = pc2

    // Workspace layout: [pack of pc2][pack of pc1][w1][w2]
    float* pack2 = (float*)d_ws;                         // B*N2*4 floats
    float* pack1 = pack2 + (size_t)B * N2 * 4;           // B*N1*4 floats
    float* w1    = pack1 + (size_t)B * N1 * 4;           // B*s1 floats
    float* w2    = w1 + (size_t)B * s1;                  // B*s2 floats

    // Pack B operands for both directions
    chamfer_pack<<<dim3((N2 * 2 + 127) / 128, B), dim3(128), 0, stream>>>(pc2, pack2, N2);
    chamfer_pack<<<dim3((N1 * 2 + 127) / 128, B), dim3(128), 0, stream>>>(pc1, pack1, N1);

    const dim3 blk(128);                        // 4 waves/block, 1 strip/wave
    chamfer_dir_wmma<<<dim3((s1 + 3) / 4, B), blk, 0, stream>>>(pc1, pack2, w1, N1, N2); // d12
    chamfer_dir_wmma<<<dim3((s2 + 3) / 4, B), blk, 0, stream>>>(pc2, pack1, w2, N2, N1); // d21
    chamfer_reduce<<<dim3(B), dim3(256), 0, stream>>>(w1, s1, w2, s2, out);
}